// BPRLossWithNoClick_2439541424724
// MI455X (gfx1250) — compile-verified
//
#include <hip/hip_runtime.h>

typedef float v2f __attribute__((ext_vector_type(2)));
typedef float v8f __attribute__((ext_vector_type(8)));

#define NB 16      // B
#define NT 100     // T
#define NV 50000   // V
#define NS 4       // S
#define TPAD 128   // T padded to multiple of K=4 (and nice LDS stride)
#define NTHREADS 512

// Numerically stable log-sigmoid matching jax.nn.log_sigmoid:
//   log_sigmoid(x) = min(x, 0) - log1p(exp(-|x|))
__device__ __forceinline__ float log_sigmoidf(float x) {
    return fminf(x, 0.0f) - log1pf(expf(-fabsf(x)));
}

__global__ __launch_bounds__(NTHREADS)
void bpr_loss_kernel(const float* __restrict__ output,
                     const int*   __restrict__ labels,
                     const int*   __restrict__ x_lens,
                     const int*   __restrict__ neg_ids,
                     float*       __restrict__ out)
{
    __shared__ float P[NB][TPAD];   // masked per-(b,t) loss, zero-padded
    __shared__ float rowsum[NB];    // per-user sums from WMMA

    const int tid = threadIdx.x;

    // ---- zero the padded loss matrix (mask padding + t >= x_lens) ----
    for (int i = tid; i < NB * TPAD; i += NTHREADS)
        (&P[0][0])[i] = 0.0f;
    __syncthreads();

    // ---- gather + elementwise: 1600 (b,t) items, 5 scattered loads each ----
    // Latency-bound: 512 threads * ~3 items each gives deep MLP into HBM.
    for (int item = tid; item < NB * NT; item += NTHREADS) {
        const int b = item / NT;
        const int t = item - b * NT;
        if (t < x_lens[b]) {
            const size_t bt  = (size_t)(b * NT + t);
            const float* row = output + bt * (size_t)NV;
            const float  pos = row[labels[bt]];
            const int*   ni  = neg_ids + bt * (size_t)NS;
            float acc = 0.0f;
#pragma unroll
            for (int s = 0; s < NS; ++s) {
                const float x = pos - row[ni[s]];
                acc += log_sigmoidf(x);
            }
            P[b][t] = acc;
        }
    }
    __syncthreads();

    // ---- row-sum reduction over T on the matrix unit ----
    // D = A(16x4) * ones(4x16) + C accumulates D[m,n] = sum_k A[m,k];
    // chained over TPAD/4 = 32 chunks -> D[m,*] = sum_t P[m][t].
    // Wave 0 only: tid<32 is wave-uniform, so EXEC is all-ones inside (WMMA req).
    if (tid < 32) {
        const int m   = tid & 15;
        const int klo = (tid < 16) ? 0 : 2;   // A layout: VGPR0=K0|K2, VGPR1=K1|K3
        v2f ones; ones.x = 1.0f; ones.y = 1.0f;
        v8f acc = {};
#pragma unroll 4
        for (int c = 0; c < TPAD / 4; ++c) {
            v2f a;
            a.x = P[m][c * 4 + klo];
            a.y = P[m][c * 4 + klo + 1];
            acc = __builtin_amdgcn_wmma_f32_16x16x4_f32(
                /*neg_a=*/false, a, /*neg_b=*/false, ones,
                /*c_mod=*/(short)0, acc, /*reuse_a=*/false, /*reuse_b=*/false);
        }
        // C/D layout: VGPR r -> M=r (lanes 0-15), M=r+8 (lanes 16-31); take N=0.
        if (tid == 0) {
#pragma unroll
            for (int r = 0; r < 8; ++r) rowsum[r] = acc[r];
        }
        if (tid == 16) {
#pragma unroll
            for (int r = 0; r < 8; ++r) rowsum[8 + r] = acc[r];
        }
    }
    __syncthreads();

    // ---- finalize: -mean_b( rowsum[b] / x_lens[b] ) ----
    if (tid == 0) {
        float total = 0.0f;
#pragma unroll
        for (int b = 0; b < NB; ++b)
            total += rowsum[b] / (float)x_lens[b];
        out[0] = -total / (float)NB;
    }
}

extern "C" void kernel_launch(void* const* d_in, const int* in_sizes, int n_in,
                              void* d_out, int out_size, void* d_ws, size_t ws_size,
                              hipStream_t stream) {
    (void)in_sizes; (void)n_in; (void)out_size; (void)d_ws; (void)ws_size;
    const float* output  = (const float*)d_in[0];  // [B,T,V] f32
    const int*   labels  = (const int*)  d_in[1];  // [B,T]   i32
    const int*   x_lens  = (const int*)  d_in[2];  // [B]     i32
    // d_in[3] = uids (unused by reference)
    const int*   neg_ids = (const int*)  d_in[4];  // [B,T,S] i32
    float*       out     = (float*)d_out;          // [1]     f32

    bpr_loss_kernel<<<1, NTHREADS, 0, stream>>>(output, labels, x_lens, neg_ids, out);
}